// GCN_42649025249306
// MI455X (gfx1250) — compile-verified
//
#include <hip/hip_runtime.h>
#include <hip/hip_bf16.h>

// GCN on MI455X (gfx1250, wave32). fp32 end-to-end; GEMMs via V_WMMA_F32_16X16X4_F32.

typedef float v2f __attribute__((ext_vector_type(2)));
typedef float v8f __attribute__((ext_vector_type(8)));

#define FEAT 128          // IN == H == 128
#define APAD 132          // LDS row pitch for A tile (bank-conflict avoidance)

// ---------------- degree / normalization ----------------

__global__ void k_deg_init(float* __restrict__ deg, int n) {
    int i = blockIdx.x * blockDim.x + threadIdx.x;
    if (i < n) deg[i] = 1.0f;                      // self-loop contributes 1
}

__global__ void k_deg_count(const int* __restrict__ rows, float* __restrict__ deg, int e) {
    int i = blockIdx.x * blockDim.x + threadIdx.x;
    if (i < e) atomicAdd(&deg[rows[i]], 1.0f);
}

__global__ void k_dinv(const float* __restrict__ deg, float* __restrict__ dinv, int n) {
    int i = blockIdx.x * blockDim.x + threadIdx.x;
    if (i < n) dinv[i] = rsqrtf(deg[i]);           // deg >= 1 always (self-loop)
}

// ---------------- WMMA GEMM: C[M x 128] = A[M x 128] * W[128 x 128] ----------------
// One block = 256 threads = 8 wave32s. Block computes a 16 x 128 output slab:
// wave w owns the 16x16 tile at columns [16w, 16w+16). K-loop: 32 steps of K=4.
//
// f32 WMMA VGPR layouts (ISA 7.12.2):
//   A (16x4):  lane<16 -> {K=k0,k0+1}, lane>=16 -> {K=k0+2,k0+3}; lane%16 = M
//   B (4x16):  lane<16 -> {K=k0,k0+1}, lane>=16 -> {K=k0+2,k0+3}; lane%16 = N (row-striped)
//   C/D (16x16): vgpr i -> M = i + 8*(lane>=16), N = lane%16
template <bool RELU_A>
__global__ __launch_bounds__(256) void k_gemm_wmma(const float* __restrict__ A,
                                                   const float* __restrict__ W,
                                                   float* __restrict__ C) {
    __shared__ float sW[FEAT * FEAT];   // 64 KB: whole weight matrix
    __shared__ float sA[16 * APAD];     // padded 16x128 A tile

    const int tid    = threadIdx.x;
    const int m_base = blockIdx.x * 16;

    // Stage W (coalesced float4): 4096 float4 / 256 threads = 16 each.
    const float4* W4  = (const float4*)W;
    float4*       sW4 = (float4*)sW;
    #pragma unroll
    for (int i = 0; i < (FEAT * FEAT / 4) / 256; ++i)
        sW4[tid + i * 256] = W4[tid + i * 256];

    // Stage A tile (scalar due to row padding), fusing ReLU for layer 2.
    const float* Ab = A + (size_t)m_base * FEAT;
    #pragma unroll
    for (int i = 0; i < (16 * FEAT) / 256; ++i) {
        int idx = tid + i * 256;
        int r = idx >> 7, c = idx & (FEAT - 1);
        float v = Ab[idx];
        if (RELU_A) v = fmaxf(v, 0.0f);
        sA[r * APAD + c] = v;
    }
    __syncthreads();

    const int wave   = tid >> 5;       // 0..7
    const int lane   = tid & 31;
    const int lrow   = lane & 15;
    const int hi     = lane >> 4;      // 0: K 0,1  1: K 2,3
    const int n_base = wave * 16;

    v8f acc = {};
    #pragma unroll
    for (int k0 = 0; k0 < FEAT; k0 += 4) {
        v2f a, b;
        const int ka = k0 + 2 * hi;
        a.x = sA[lrow * APAD + ka];
        a.y = sA[lrow * APAD + ka + 1];
        b.x = sW[(ka)     * FEAT + n_base + lrow];
        b.y = sW[(ka + 1) * FEAT + n_base + lrow];
        acc = __builtin_amdgcn_wmma_f32_16x16x4_f32(
            /*neg_a=*/false, a, /*neg_b=*/false, b,
            /*c_mod=*/(short)0, acc, /*reuse_a=*/false, /*reuse_b=*/false);
    }

    float* Crow = C + (size_t)m_base * FEAT + n_base + lrow;
    #pragma unroll
    for (int i = 0; i < 8; ++i)
        Crow[(size_t)(i + 8 * hi) * FEAT] = acc[i];
}

// ---------------- self-loop + bias init: out = dinv[i]^2 * xw + b ----------------

__global__ void k_init_out(const float* __restrict__ xw, const float* __restrict__ dinv,
                           const float* __restrict__ b, float* __restrict__ out, int n) {
    int t = blockIdx.x * blockDim.x + threadIdx.x;      // n*32 threads, float4 each
    if (t >= n * (FEAT / 4)) return;
    int i  = t >> 5;
    int f4 = (t & 31) * 4;
    float s = dinv[i];
    s = s * s;
    float4 v  = *(const float4*)(xw + (size_t)i * FEAT + f4);
    float4 bb = *(const float4*)(b + f4);
    float4 o  = make_float4(s * v.x + bb.x, s * v.y + bb.y,
                            s * v.z + bb.z, s * v.w + bb.w);
    *(float4*)(out + (size_t)i * FEAT + f4) = o;
}

// ---------------- edge scatter: out[row] += dinv[row]*dinv[col] * xw[col] ----------------
// 32 threads per edge, float4 per thread, f32 global atomics (xw is L2-resident: 25.6MB << 192MB).

__global__ void k_scatter(const int* __restrict__ rows, const int* __restrict__ cols,
                          const float* __restrict__ dinv, const float* __restrict__ xw,
                          float* __restrict__ out, int e) {
    int t = blockIdx.x * blockDim.x + threadIdx.x;
    int edge = t >> 5;
    if (edge >= e) return;
    int f4 = (t & 31) * 4;
    int r = rows[edge];
    int c = cols[edge];
    float nrm = dinv[r] * dinv[c];
    float4 v = *(const float4*)(xw + (size_t)c * FEAT + f4);
    float* o = out + (size_t)r * FEAT + f4;
    atomicAdd(o + 0, nrm * v.x);
    atomicAdd(o + 1, nrm * v.y);
    atomicAdd(o + 2, nrm * v.z);
    atomicAdd(o + 3, nrm * v.w);
}

// ---------------- launcher ----------------

extern "C" void kernel_launch(void* const* d_in, const int* in_sizes, int n_in,
                              void* d_out, int out_size, void* d_ws, size_t ws_size,
                              hipStream_t stream) {
    const float* x  = (const float*)d_in[0];
    const int*   ei = (const int*)d_in[1];
    const float* W1 = (const float*)d_in[2];
    const float* b1 = (const float*)d_in[3];
    const float* W2 = (const float*)d_in[4];
    const float* b2 = (const float*)d_in[5];
    float* out = (float*)d_out;

    const int N = in_sizes[0] / FEAT;   // 50000
    const int E = in_sizes[1] / 2;      // 800000
    const int* rows = ei;
    const int* cols = ei + E;

    // Workspace layout (floats): deg[N] | dinv[N] | xw[N*128] | h1[N*128]
    float* deg  = (float*)d_ws;
    float* dinv = deg + N;
    float* xw   = dinv + N;             // reused for both layers
    float* h1   = xw + (size_t)N * FEAT;

    const int TB = 256;
    // normalization
    k_deg_init<<<(N + TB - 1) / TB, TB, 0, stream>>>(deg, N);
    k_deg_count<<<(E + TB - 1) / TB, TB, 0, stream>>>(rows, deg, E);
    k_dinv<<<(N + TB - 1) / TB, TB, 0, stream>>>(deg, dinv, N);

    const int gemm_blocks = N / 16;                     // 50000 % 16 == 0
    const int init_blocks = (N * (FEAT / 4) + TB - 1) / TB;
    const int scat_blocks = ((E * 32) + TB - 1) / TB;

    // layer 1
    k_gemm_wmma<false><<<gemm_blocks, TB, 0, stream>>>(x, W1, xw);
    k_init_out<<<init_blocks, TB, 0, stream>>>(xw, dinv, b1, h1, N);
    k_scatter<<<scat_blocks, TB, 0, stream>>>(rows, cols, dinv, xw, h1, E);

    // layer 2 (ReLU fused into A staging)
    k_gemm_wmma<true><<<gemm_blocks, TB, 0, stream>>>(h1, W2, xw);
    k_init_out<<<init_blocks, TB, 0, stream>>>(xw, dinv, b2, out, N);
    k_scatter<<<scat_blocks, TB, 0, stream>>>(rows, cols, dinv, xw, out, E);
}